// PtModule_73667279061132
// MI455X (gfx1250) — compile-verified
//
#include <hip/hip_runtime.h>
#include <hip/hip_bf16.h>

// out[i,j] = 2*x[i,j] + 5 - (i + j), x: 8192x8192 fp32.
// Pure streaming kernel: 512 MB traffic, ~22 us floor at 23.3 TB/s.
// Strategy: float4 (B128) nontemporal loads/stores, grid-stride loop,
// wave32-friendly 256-thread blocks.

typedef float v4f __attribute__((ext_vector_type(4)));

__global__ __launch_bounds__(256) void fused_outer_sum_kernel(
    const v4f* __restrict__ x, v4f* __restrict__ out, int total4, int row_shift,
    int col_mask) {
  int idx = blockIdx.x * blockDim.x + threadIdx.x;
  const int stride = gridDim.x * blockDim.x;

  for (; idx < total4; idx += stride) {
    // Nontemporal B128 load: single-touch data, don't pollute L2 (TH=NT).
    v4f v = __builtin_nontemporal_load(&x[idx]);

    // Element index of lane's first float; 4 consecutive elements share a row
    // since the row length (8192) is a multiple of 4.
    const int e = idx << 2;
    const int i = e >> row_shift;   // row
    const int j = e & col_mask;     // col of first element
    const float c = (float)(i + j); // exact: max 16382 << 2^24

    v4f r;
    r.x = v.x * 2.0f + 5.0f - c;
    r.y = v.y * 2.0f + 5.0f - (c + 1.0f);
    r.z = v.z * 2.0f + 5.0f - (c + 2.0f);
    r.w = v.w * 2.0f + 5.0f - (c + 3.0f);

    // Nontemporal B128 store.
    __builtin_nontemporal_store(r, &out[idx]);
  }
}

extern "C" void kernel_launch(void* const* d_in, const int* in_sizes, int n_in,
                              void* d_out, int out_size, void* d_ws, size_t ws_size,
                              hipStream_t stream) {
  (void)n_in; (void)d_ws; (void)ws_size;

  const float* x = (const float*)d_in[0];
  float* out = (float*)d_out;

  // 8192 x 8192 fp32 (row-major). Derive sizes from the harness-provided count.
  const int total = in_sizes[0];          // 67,108,864 elements
  const int total4 = total >> 2;          // 16,777,216 float4
  const int M = 8192;                     // row length (power of two)
  const int row_shift = 13;               // log2(M)
  const int col_mask = M - 1;

  // ~8 float4 per thread: 8192 blocks x 256 threads (= 8 wave32/block) keeps
  // every WGP saturated with outstanding loads while bounding index overhead.
  const int threads = 256;
  int blocks = (total4 + threads * 8 - 1) / (threads * 8);
  if (blocks < 1) blocks = 1;

  fused_outer_sum_kernel<<<blocks, threads, 0, stream>>>(
      (const v4f*)x, (v4f*)out, total4, row_shift, col_mask);

  (void)out_size;
}